// GradientOutput_1623497637929
// MI455X (gfx1250) — compile-verified
//
#include <hip/hip_runtime.h>

// ---------------------------------------------------------------------------
// GradientOutput for MI455X (gfx1250, wave32).
//
// Roofline: ~205 MB streamed once -> ~9 us at 23.3 TB/s; ~0.13 GFLOP total.
// Memory/atomic bound. Strategy:
//   * 4 edges per thread -> 16B-aligned streams -> global_load_b128 NT loads
//     (edge_diff/dE: 3x b128 each, edge_idx: 2x b128 per group).
//   * forces: global_atomic_add_f32 (no-return), hot 2.4 MB array L2-resident.
//   * virial: fused atom->image segment sum; per-block LDS accumulators
//     (ds_add_f32), flushed into 16 partial buffers in d_ws.
//   * Final 16-way partial reduction on the tensor core:
//     V_WMMA_F32_16X16X4_F32 with A = ones(16x4); D row 0 = column sums.
//     Guard-free main loop so EXEC stays all-ones through the WMMA region.
// ---------------------------------------------------------------------------

typedef __attribute__((ext_vector_type(2))) float v2f;
typedef __attribute__((ext_vector_type(4))) float v4f;
typedef __attribute__((ext_vector_type(8))) float v8f;
typedef __attribute__((ext_vector_type(4))) int   v4i;

#define NPART      16      // partial virial buffers (one 16-deep WMMA reduction)
#define MAX_IMAGES 128

__global__ void zero_kernel(float* __restrict__ forces, float* __restrict__ ws,
                            int n_forces, int n_ws) {
    int i = blockIdx.x * blockDim.x + threadIdx.x;
    if (i < n_forces) forces[i] = 0.0f;
    if (i < n_ws)     ws[i]     = 0.0f;
}

__global__ void edge_scatter_kernel(const float* __restrict__ edge_diff,
                                    const float* __restrict__ dE,
                                    const int*   __restrict__ edge_idx,
                                    const int*   __restrict__ image_idx,
                                    float* __restrict__ forces,
                                    float* __restrict__ part_virial,
                                    int n_edges, int n_images)
{
    __shared__ float vloc[MAX_IMAGES * 9];   // 4.6 KB of the 320 KB WGP LDS
    const int tid  = threadIdx.x;
    const int nAcc = n_images * 9;

    for (int t = tid; t < nAcc; t += blockDim.x) vloc[t] = 0.0f;
    __syncthreads();

    const int n_groups = (n_edges + 3) >> 2;           // 4 edges per thread
    const int stride   = gridDim.x * blockDim.x;

    for (int grp = blockIdx.x * blockDim.x + tid; grp < n_groups; grp += stride) {
        const int e0 = grp << 2;

        // 16B-aligned non-temporal wide loads (read-once streams, keep L2
        // free for the atomic-hot forces array).
        union { v4f v[3]; float f[12]; } dbuf, gbuf;
        union { v4i v[2]; int   s[8];  } ibuf;
        const v4f* dp = (const v4f*)(edge_diff + 3 * e0);
        const v4f* gp = (const v4f*)(dE        + 3 * e0);
        const v4i* ip = (const v4i*)(edge_idx  + 2 * e0);
        dbuf.v[0] = __builtin_nontemporal_load(dp + 0);
        dbuf.v[1] = __builtin_nontemporal_load(dp + 1);
        dbuf.v[2] = __builtin_nontemporal_load(dp + 2);
        gbuf.v[0] = __builtin_nontemporal_load(gp + 0);
        gbuf.v[1] = __builtin_nontemporal_load(gp + 1);
        gbuf.v[2] = __builtin_nontemporal_load(gp + 2);
        ibuf.v[0] = __builtin_nontemporal_load(ip + 0);
        ibuf.v[1] = __builtin_nontemporal_load(ip + 1);

        const int nk = (e0 + 4 <= n_edges) ? 4 : (n_edges - e0);
#pragma unroll
        for (int k = 0; k < 4; ++k) {
            if (k >= nk) break;
            const int   i  = ibuf.s[2 * k + 0];
            const int   j  = ibuf.s[2 * k + 1];
            const float dx = dbuf.f[3 * k + 0];
            const float dy = dbuf.f[3 * k + 1];
            const float dz = dbuf.f[3 * k + 2];
            const float gx = gbuf.f[3 * k + 0];
            const float gy = gbuf.f[3 * k + 1];
            const float gz = gbuf.f[3 * k + 2];

            // forces: i_forces(+) and j_forces(-) fused
            unsafeAtomicAdd(&forces[3 * i + 0],  gx);
            unsafeAtomicAdd(&forces[3 * i + 1],  gy);
            unsafeAtomicAdd(&forces[3 * i + 2],  gz);
            unsafeAtomicAdd(&forces[3 * j + 0], -gx);
            unsafeAtomicAdd(&forces[3 * j + 1], -gy);
            unsafeAtomicAdd(&forces[3 * j + 2], -gz);

            // virial: fused atom->image segment sum, block-local in LDS
            const int img = image_idx[i];       // 800 KB table, cache-resident
            float* vb = &vloc[img * 9];
            atomicAdd(vb + 0, dx * gx);
            atomicAdd(vb + 1, dx * gy);
            atomicAdd(vb + 2, dx * gz);
            atomicAdd(vb + 3, dy * gx);
            atomicAdd(vb + 4, dy * gy);
            atomicAdd(vb + 5, dy * gz);
            atomicAdd(vb + 6, dz * gx);
            atomicAdd(vb + 7, dz * gy);
            atomicAdd(vb + 8, dz * gz);
        }
    }
    __syncthreads();

    // flush block accumulator into one of NPART partial buffers
    float* dst = part_virial + (blockIdx.x & (NPART - 1)) * nAcc;
    for (int t = tid; t < nAcc; t += blockDim.x)
        unsafeAtomicAdd(&dst[t], vloc[t]);
}

// One wave32. Reduces NPART partial virials with V_WMMA_F32_16X16X4_F32 and
// writes virial + stress (= -virial / det(cell)).
__global__ void finish_kernel(const float* __restrict__ part_virial,
                              const float* __restrict__ cell,
                              float* __restrict__ virial_out,
                              float* __restrict__ stress_out,
                              int n_images)
{
    __shared__ float vol[MAX_IMAGES];
    const int lane = threadIdx.x;            // 0..31, EXEC all ones for WMMA

    // volumes: vol = row0 . (row1 x row2)
    for (int img = lane; img < n_images; img += 32) {
        const float* c = &cell[img * 9];
        const float a0 = c[0], a1 = c[1], a2 = c[2];
        const float b0 = c[3], b1 = c[4], b2 = c[5];
        const float c0 = c[6], c1 = c[7], c2 = c[8];
        vol[img] = a0 * (b1 * c2 - b2 * c1)
                 + a1 * (b2 * c0 - b0 * c2)
                 + a2 * (b0 * c1 - b1 * c0);
    }
    __syncthreads();

    const int total   = n_images * 9;        // 1152 for the reference shapes
    const int nchunks = total >> 4;          // guard-free WMMA main loop

    // A = ones(16x4): D[m][n] = sum_k B[k][n] for every row m.
    v2f a; a.x = 1.0f; a.y = 1.0f;
    // B layout (4x16 f32): VGPR0 lanes0-15 = K0, lanes16-31 = K2;
    //                      VGPR1 lanes0-15 = K1, lanes16-31 = K3.
    const int halfSel = (lane < 16) ? 0 : 2;
    const int col     = (lane & 15);

    for (int chunk = 0; chunk < nchunks; ++chunk) {
        const float* pv = part_virial + halfSel * total + (chunk << 4) + col;
        v8f acc = {0.0f, 0.0f, 0.0f, 0.0f, 0.0f, 0.0f, 0.0f, 0.0f};
#pragma unroll
        for (int w = 0; w < NPART / 4; ++w) {
            v2f b;
            b.x = pv[(4 * w + 0) * total];
            b.y = pv[(4 * w + 1) * total];
            acc = __builtin_amdgcn_wmma_f32_16x16x4_f32(
                      /*neg_a=*/false, a, /*neg_b=*/false, b,
                      /*c_mod=*/(short)0, acc, /*reuse_a=*/false, /*reuse_b=*/false);
        }
        if (lane < 16) {
            const int o = (chunk << 4) + lane; // D row M=0: VGPR0, lanes 0-15
            const float s = acc[0];
            virial_out[o] = s;
            stress_out[o] = -s / vol[o / 9];
        }
    }

    // scalar tail if total is not a multiple of 16 (not hit for 128 images)
    for (int o = (nchunks << 4) + lane; o < total; o += 32) {
        float s = 0.0f;
        for (int p = 0; p < NPART; ++p) s += part_virial[p * total + o];
        virial_out[o] = s;
        stress_out[o] = -s / vol[o / 9];
    }
}

extern "C" void kernel_launch(void* const* d_in, const int* in_sizes, int n_in,
                              void* d_out, int out_size, void* d_ws, size_t ws_size,
                              hipStream_t stream)
{
    const float* edge_diff = (const float*)d_in[0];
    const float* dE        = (const float*)d_in[1];
    const float* cell      = (const float*)d_in[2];
    const int*   edge_idx  = (const int*)d_in[3];
    const int*   image_idx = (const int*)d_in[4];

    const int n_edges  = in_sizes[0] / 3;
    const int n_images = in_sizes[2] / 9;
    const int n_atoms  = in_sizes[4];

    float* out        = (float*)d_out;
    float* forces     = out;                         // 3 * n_atoms
    float* virial_out = out + 3 * n_atoms;           // n_images * 9
    float* stress_out = virial_out + n_images * 9;   // n_images * 9

    float* part_virial = (float*)d_ws;               // NPART * n_images * 9 floats

    const int n_forces = 3 * n_atoms;
    const int n_wsf    = NPART * n_images * 9;
    const int nz       = (n_forces > n_wsf) ? n_forces : n_wsf;
    zero_kernel<<<(nz + 255) / 256, 256, 0, stream>>>(forces, part_virial,
                                                      n_forces, n_wsf);

    // 2048 blocks x 256 threads (8 wave32/block); 4 edges/thread/iteration,
    // ~3 grid-stride iterations for 6.4M edges.
    edge_scatter_kernel<<<2048, 256, 0, stream>>>(edge_diff, dE, edge_idx,
                                                  image_idx, forces, part_virial,
                                                  n_edges, n_images);

    finish_kernel<<<1, 32, 0, stream>>>(part_virial, cell, virial_out,
                                        stress_out, n_images);
}